// SkipGram_77867757076987
// MI455X (gfx1250) — compile-verified
//
#include <hip/hip_runtime.h>

typedef __attribute__((ext_vector_type(2))) float v2f;
typedef __attribute__((ext_vector_type(8))) float v8f;

#define DIMSZ 128          // embedding dim (32 WMMA k-steps of 4)
#define GROUP 16           // n-rows per wave (WMMA tile M/N)
#define WAVES_PER_BLOCK 8

// z[n,k] = sum_d (U_w[d, context[n,k]] + U_b[d]) * (V_w[d, center[n]] + V_b[d])
// One wave32 handles 16 consecutive n. B tile = 16 center columns (+bias),
// A tile = 16 context columns (+bias) for a given k. C = A x B^ via 32 chained
// v_wmma_f32_16x16x4_f32; diagonal C[i,i] is the wanted score (f32-exact).
__global__ __launch_bounds__(WAVES_PER_BLOCK * 32)
void skipgram_scores_wmma(const int* __restrict__ center,    // [N]
                          const int* __restrict__ context,   // [N,K]
                          const float* __restrict__ U_w,     // [DIM, VOCAB]
                          const float* __restrict__ U_b,     // [DIM]
                          const float* __restrict__ V_w,     // [DIM, VOCAB]
                          const float* __restrict__ V_b,     // [DIM]
                          float* __restrict__ out,           // [N, K]
                          int N, int K, int vocab)
{
    const int lane = threadIdx.x & 31;
    const int wave = threadIdx.x >> 5;
    const int group = blockIdx.x * WAVES_PER_BLOCK + wave;
    const int n_base = group * GROUP;
    if (n_base >= N) return;                 // whole-wave exit only (N%16==0)

    const int row  = lane & 15;              // i (A row) / j (B col) this lane owns
    const int half = lane >> 4;              // selects K-slot pair {0,1} vs {2,3}
    int n_row = n_base + row;
    if (n_row >= N) n_row = N - 1;           // clamp (keeps EXEC all-ones)

    // ---- B fragments: v[n_j] + V_b, resident for all k ----
    const int cidx = center[n_row];
    v2f   b[DIMSZ / 4];
    float ub0[DIMSZ / 4], ub1[DIMSZ / 4];
    #pragma unroll
    for (int s = 0; s < DIMSZ / 4; ++s) {
        const int d0 = 4 * s + 2 * half;
        b[s].x = V_w[(size_t)d0       * vocab + cidx] + V_b[d0];
        b[s].y = V_w[(size_t)(d0 + 1) * vocab + cidx] + V_b[d0 + 1];
        ub0[s] = U_b[d0];
        ub1[s] = U_b[d0 + 1];
    }

    // diagonal ownership: i in 0..7 -> lane i, c[i];  i in 8..15 -> lane i+16, c[i-8]
    const bool writes = (lane < 8) || (lane >= 24);
    const int  i_out  = (lane < 8) ? lane : (lane - 16);
    const int  comp   = (lane < 8) ? lane : (lane - 24);

    for (int k = 0; k < K; ++k) {
        const int uidx = context[(size_t)n_row * K + k];
        v8f c = {};
        #pragma unroll
        for (int s = 0; s < DIMSZ / 4; ++s) {
            const int d0 = 4 * s + 2 * half;
            v2f a;
            a.x = U_w[(size_t)d0       * vocab + uidx] + ub0[s];
            a.y = U_w[(size_t)(d0 + 1) * vocab + uidx] + ub1[s];
            // 8 args: (neg_a, A, neg_b, B, c_mod, C, reuse_a, reuse_b)
            c = __builtin_amdgcn_wmma_f32_16x16x4_f32(false, a, false, b[s],
                                                      (short)0, c, false, false);
        }
        // extract this lane's diagonal component (small cndmask chain)
        float diag = c[0];
        diag = (comp == 1) ? c[1] : diag;
        diag = (comp == 2) ? c[2] : diag;
        diag = (comp == 3) ? c[3] : diag;
        diag = (comp == 4) ? c[4] : diag;
        diag = (comp == 5) ? c[5] : diag;
        diag = (comp == 6) ? c[6] : diag;
        diag = (comp == 7) ? c[7] : diag;

        const int n_out = n_base + i_out;
        if (writes && n_out < N)
            out[(size_t)n_out * K + k] = diag;
    }
}

extern "C" void kernel_launch(void* const* d_in, const int* in_sizes, int n_in,
                              void* d_out, int out_size, void* d_ws, size_t ws_size,
                              hipStream_t stream) {
    (void)n_in; (void)d_ws; (void)ws_size;
    const int*   center  = (const int*)  d_in[0];   // [N]
    const int*   context = (const int*)  d_in[1];   // [N,K]
    const float* U_w     = (const float*)d_in[2];   // [DIM,VOCAB]
    const float* U_b     = (const float*)d_in[3];   // [DIM]
    const float* V_w     = (const float*)d_in[4];   // [DIM,VOCAB]
    const float* V_b     = (const float*)d_in[5];   // [DIM]
    float*       out     = (float*)d_out;           // [N,K] f32

    const int N     = in_sizes[0];
    const int K     = in_sizes[1] / N;              // 10
    const int dim   = in_sizes[3];                  // 128 (compile-time DIMSZ)
    const int vocab = in_sizes[2] / dim;            // 100000
    (void)out_size;

    const int groups = (N + GROUP - 1) / GROUP;
    const int blocks = (groups + WAVES_PER_BLOCK - 1) / WAVES_PER_BLOCK;
    skipgram_scores_wmma<<<blocks, WAVES_PER_BLOCK * 32, 0, stream>>>(
        center, context, U_w, U_b, V_w, V_b, out, N, K, vocab);
}